// mk1_Encoder_49976239456542
// MI455X (gfx1250) — compile-verified
//
#include <hip/hip_runtime.h>
#include <math.h>

#define N_NODES 50000
#define N_EDGES 600000
#define IN_CH   128
#define HID     128
#define ENC_H   100
#define OUT_CH  32
#define N_CODES 40
#define AA_DIM  20
#define LN_EPS  1e-5f

// GEMM block tiling
#define BM 128
#define BN 32
#define BK 32
#define LSTR 36          // LDS row stride (floats): even (8B-aligned v2f) + conflict-free b64 reads

typedef float v2f __attribute__((ext_vector_type(2)));
typedef float v8f __attribute__((ext_vector_type(8)));

__device__ __forceinline__ float gelu_exact(float x) {
  return 0.5f * x * (1.0f + erff(x * 0.70710678118654752f));
}

__device__ __forceinline__ float wave_sum32(float s) {
#pragma unroll
  for (int off = 16; off > 0; off >>= 1) s += __shfl_xor(s, off, 32);
  return s;
}

// ---------------- double LayerNorm (row = 128 ch), one wave32 per row ----------------
__global__ void mk1_ln2_kernel(const float* __restrict__ x,
                               const float* __restrict__ g1, const float* __restrict__ b1,
                               const float* __restrict__ g2, const float* __restrict__ b2,
                               float* __restrict__ out) {
  int w    = (blockIdx.x * blockDim.x + threadIdx.x) >> 5;
  int lane = threadIdx.x & 31;
  if (w >= N_NODES) return;
  float4 v = ((const float4*)(x + (size_t)w * IN_CH))[lane];

  float m  = wave_sum32(v.x + v.y + v.z + v.w) * (1.0f / 128.0f);
  float dx = v.x - m, dy = v.y - m, dz = v.z - m, dw = v.w - m;
  float rs = rsqrtf(wave_sum32(dx*dx + dy*dy + dz*dz + dw*dw) * (1.0f/128.0f) + LN_EPS);
  float4 G = ((const float4*)g1)[lane], B = ((const float4*)b1)[lane];
  float4 y;
  y.x = G.x*dx*rs + B.x; y.y = G.y*dy*rs + B.y; y.z = G.z*dz*rs + B.z; y.w = G.w*dw*rs + B.w;

  m  = wave_sum32(y.x + y.y + y.z + y.w) * (1.0f / 128.0f);
  dx = y.x - m; dy = y.y - m; dz = y.z - m; dw = y.w - m;
  rs = rsqrtf(wave_sum32(dx*dx + dy*dy + dz*dz + dw*dw) * (1.0f/128.0f) + LN_EPS);
  G = ((const float4*)g2)[lane]; B = ((const float4*)b2)[lane];
  float4 o;
  o.x = G.x*dx*rs + B.x; o.y = G.y*dy*rs + B.y; o.z = G.z*dz*rs + B.z; o.w = G.w*dw*rs + B.w;
  ((float4*)(out + (size_t)w * IN_CH))[lane] = o;
}

// ---------------- LDS-staged fp32 WMMA GEMM ----------------
// D = [C +] A(MxK) @ B(KxN) + bias, act. Block = 256 thr = 8 waves -> 128x32 tile.
// Waves 4(M) x 2(N); each wave: 32x16 via two 16x16 WMMA accumulators.
// act: 0=none 1=gelu 2=tanh ; accumulate: D += existing C contents
__global__ __launch_bounds__(256)
void mk2_gemm_wmma_f32(const float* __restrict__ A, const float* __restrict__ B,
                       float* __restrict__ C, const float* __restrict__ bias,
                       int M, int N, int K, int lda, int ldb, int ldc,
                       int accumulate, int act) {
  __shared__ float sA[BM * LSTR];   // 18432 B, sA[row][k]
  __shared__ float sB[BN * LSTR];   //  4608 B, transposed: sB[n][k]

  const int t    = threadIdx.x;
  const int lane = t & 31;
  const int wave = t >> 5;
  const int l15  = lane & 15;
  const int koff = (lane >> 4) << 1;         // 0 or 2 per ISA A/B lane layout
  const int wn   = wave & 1;
  const int wm   = wave >> 1;

  const int nbn      = (N + BN - 1) / BN;
  const int bm       = blockIdx.x / nbn;
  const int bn       = blockIdx.x - bm * nbn;
  const int row_base = bm * BM;
  const int col_base = bn * BN;

  v8f acc0 = {0.f,0.f,0.f,0.f,0.f,0.f,0.f,0.f};
  v8f acc1 = {0.f,0.f,0.f,0.f,0.f,0.f,0.f,0.f};

  for (int kc = 0; kc < K; kc += BK) {
    // ---- stage A tile (BM x BK), coalesced float4 ----
#pragma unroll
    for (int i = 0; i < 4; ++i) {
      int id = t + 256 * i;                  // 0..1023 -> 128 rows x 8 quads
      int r  = id >> 3;
      int q  = (id & 7) << 2;
      int gr = row_base + r; if (gr >= M) gr = M - 1;   // clamp (stores guarded later)
      int gk = kc + q;
      const float* ap = A + (size_t)gr * lda;
      float4 v;
      if (gk + 3 < K) {
        v = *(const float4*)(ap + gk);
      } else {
        v.x = (gk + 0 < K) ? ap[gk + 0] : 0.0f;
        v.y = (gk + 1 < K) ? ap[gk + 1] : 0.0f;
        v.z = (gk + 2 < K) ? ap[gk + 2] : 0.0f;
        v.w = (gk + 3 < K) ? ap[gk + 3] : 0.0f;
      }
      float* dst = &sA[r * LSTR + q];
      dst[0] = v.x; dst[1] = v.y; dst[2] = v.z; dst[3] = v.w;
    }
    // ---- stage B tile (BK x BN) transposed into sB[n][k], coalesced along N ----
#pragma unroll
    for (int i = 0; i < 4; ++i) {
      int id = t + 256 * i;                  // 0..1023 -> 32 k x 32 n
      int n  = id & 31;
      int k  = id >> 5;
      int gn = col_base + n;
      int gk = kc + k;
      float v = 0.0f;
      if (gn < N && gk < K) v = B[(size_t)gk * ldb + gn];
      sB[n * LSTR + k] = v;
    }
    __syncthreads();

    const float* pa0 = &sA[(wm * 32 + l15) * LSTR + koff];
    const float* pa1 = pa0 + 16 * LSTR;
    const float* pb  = &sB[(wn * 16 + l15) * LSTR + koff];
#pragma unroll
    for (int k = 0; k < BK; k += 4) {
      v2f a0 = *(const v2f*)(pa0 + k);
      v2f a1 = *(const v2f*)(pa1 + k);
      v2f b  = *(const v2f*)(pb  + k);
      acc0 = __builtin_amdgcn_wmma_f32_16x16x4_f32(false, a0, false, b, (short)0, acc0,
                                                   false, false);
      acc1 = __builtin_amdgcn_wmma_f32_16x16x4_f32(false, a1, false, b, (short)0, acc1,
                                                   false, false);
    }
    __syncthreads();
  }

  // ---- epilogue: bias + optional accumulate + activation, guarded stores ----
  const int  ncol = col_base + wn * 16 + l15;
  const bool nval = ncol < N;
  float bv = (bias != nullptr && nval) ? bias[ncol] : 0.0f;
#pragma unroll
  for (int r = 0; r < 8; ++r) {
    int m0 = row_base + wm * 32 + r + ((lane >> 4) << 3);
    int m1 = m0 + 16;
    if (nval && m0 < M) {
      float v = acc0[r] + bv;
      float* p = C + (size_t)m0 * ldc + ncol;
      if (accumulate) v += *p;
      if (act == 1) v = gelu_exact(v);
      else if (act == 2) v = tanhf(v);
      *p = v;
    }
    if (nval && m1 < M) {
      float v = acc1[r] + bv;
      float* p = C + (size_t)m1 * ldc + ncol;
      if (accumulate) v += *p;
      if (act == 1) v = gelu_exact(v);
      else if (act == 2) v = tanhf(v);
      *p = v;
    }
  }
}

// ---------------- edge scatter: one wave per edge, gather 512B row, atomic add ----------------
__global__ void mk1_scatter_kernel(const int* __restrict__ ei, const float* __restrict__ x,
                                   float* __restrict__ agg, float* __restrict__ cnt) {
  int w    = (blockIdx.x * blockDim.x + threadIdx.x) >> 5;
  int lane = threadIdx.x & 31;
  if (w >= N_EDGES) return;
  int s = ei[w];
  int d = ei[N_EDGES + w];
  float4 v = ((const float4*)(x + (size_t)s * HID))[lane];
  float* ap = agg + (size_t)d * HID + lane * 4;
  atomicAdd(ap + 0, v.x);
  atomicAdd(ap + 1, v.y);
  atomicAdd(ap + 2, v.z);
  atomicAdd(ap + 3, v.w);
  if (lane == 0) atomicAdd(cnt + d, 1.0f);
}

__global__ void mk1_agg_norm_kernel(float* __restrict__ agg, const float* __restrict__ cnt) {
  int stride = gridDim.x * blockDim.x;
  for (int i = blockIdx.x * blockDim.x + threadIdx.x; i < N_NODES * HID; i += stride) {
    float c = cnt[i >> 7];
    agg[i] *= (c > 1.0f) ? (1.0f / c) : 1.0f;
  }
}

__global__ void mk1_mean_gelu_kernel(const float* __restrict__ acc, float* __restrict__ x) {
  int stride = gridDim.x * blockDim.x;
  for (int i = blockIdx.x * blockDim.x + threadIdx.x; i < N_NODES * HID; i += stride)
    x[i] = gelu_exact(acc[i] * (1.0f / 3.0f));
}

// ---------------- GraphNorm: column sums (blockDim.x == 128, coalesced) ----------------
__global__ void mk1_colsum_kernel(const float* __restrict__ x, float* __restrict__ colsum,
                                  int rows_per_block) {
  int c  = threadIdx.x;
  int r0 = blockIdx.x * rows_per_block;
  int r1 = r0 + rows_per_block; if (r1 > N_NODES) r1 = N_NODES;
  float s = 0.0f;
  for (int r = r0; r < r1; ++r) s += x[(size_t)r * HID + c];
  atomicAdd(colsum + c, s);
}

__global__ void mk1_gn_center_kernel(const float* __restrict__ x, const float* __restrict__ colsum,
                                     const float* __restrict__ a, float* __restrict__ t,
                                     float* __restrict__ colsum2, int rows_per_block) {
  int c  = threadIdx.x;
  float am = a[c] * (colsum[c] * (1.0f / (float)N_NODES));
  int r0 = blockIdx.x * rows_per_block;
  int r1 = r0 + rows_per_block; if (r1 > N_NODES) r1 = N_NODES;
  float s2 = 0.0f;
  for (int r = r0; r < r1; ++r) {
    float v = x[(size_t)r * HID + c] - am;
    t[(size_t)r * HID + c] = v;
    s2 = fmaf(v, v, s2);
  }
  atomicAdd(colsum2 + c, s2);
}

__global__ void mk1_gn_final_kernel(const float* __restrict__ t, const float* __restrict__ colsum2,
                                    const float* __restrict__ g, const float* __restrict__ b,
                                    float* __restrict__ xout, float* __restrict__ save, int ld_save) {
  int stride = gridDim.x * blockDim.x;
  for (int i = blockIdx.x * blockDim.x + threadIdx.x; i < N_NODES * HID; i += stride) {
    int c = i & (HID - 1);
    int r = i >> 7;
    float rs = rsqrtf(colsum2[c] * (1.0f / (float)N_NODES) + LN_EPS);
    float v  = g[c] * t[i] * rs + b[c];
    xout[i] = v;
    save[(size_t)r * ld_save + c] = v;
  }
}

__global__ void mk1_concat_aa_kernel(const float* __restrict__ aa, float* __restrict__ dst) {
  int stride = gridDim.x * blockDim.x;
  for (int i = blockIdx.x * blockDim.x + threadIdx.x; i < N_NODES * AA_DIM; i += stride) {
    int r = i / AA_DIM;
    int c = i - r * AA_DIM;
    dst[(size_t)r * 128 + ENC_H + c] = aa[i];
  }
}

// ---------------- VQ: argmin over codebook, in-place q, 1.25*MSE accumulation ----------------
__global__ void mk1_vq_kernel(float* __restrict__ xq, const float* __restrict__ cb,
                              float* __restrict__ loss) {
  int i = blockIdx.x * blockDim.x + threadIdx.x;
  float se = 0.0f;
  if (i < N_NODES) {
    float xv[OUT_CH];
    float* row = xq + (size_t)i * OUT_CH;
#pragma unroll
    for (int c = 0; c < OUT_CH; ++c) xv[c] = row[c];
    float best = 3.4e38f; int bk = 0;
    for (int k = 0; k < N_CODES; ++k) {
      const float* ck = cb + k * OUT_CH;
      float d = 0.0f;
#pragma unroll
      for (int c = 0; c < OUT_CH; ++c) { float t = xv[c] - ck[c]; d = fmaf(t, t, d); }
      if (d < best) { best = d; bk = k; }
    }
    const float* ck = cb + bk * OUT_CH;
#pragma unroll
    for (int c = 0; c < OUT_CH; ++c) {
      float q = ck[c];
      float t = q - xv[c];
      se = fmaf(t, t, se);
      row[c] = q;                     // z_quantized == q in forward
    }
  }
  se = wave_sum32(se);
  if ((threadIdx.x & 31) == 0) atomicAdd(loss, se);
}

__global__ void mk1_loss_final_kernel(const float* __restrict__ loss, float* __restrict__ out) {
  if (blockIdx.x == 0 && threadIdx.x == 0)
    out[0] = 1.25f * loss[0] / (float)(N_NODES * OUT_CH);   // q_latent + COMMIT*e_latent
}

// ================= host side =================
static inline void launch_gemm(const float* A, const float* B, float* C, const float* bias,
                               int M, int N, int K, int lda, int ldb, int ldc,
                               int accum, int act, hipStream_t s) {
  int nbm = (M + BM - 1) / BM;
  int nbn = (N + BN - 1) / BN;
  mk2_gemm_wmma_f32<<<nbm * nbn, 256, 0, s>>>(A, B, C, bias, M, N, K, lda, ldb, ldc, accum, act);
}

extern "C" void kernel_launch(void* const* d_in, const int* in_sizes, int n_in,
                              void* d_out, int out_size, void* d_ws, size_t ws_size,
                              hipStream_t stream) {
  (void)in_sizes; (void)n_in; (void)out_size; (void)ws_size;
#define PF(i) ((const float*)d_in[(i)])
  const float* x_res = PF(0);
  const float* aa    = PF(1);
  const int* edges[3] = { (const int*)d_in[2], (const int*)d_in[3], (const int*)d_in[4] };
  const float* ln1_g = PF(5),  *ln1_b = PF(6);
  const float* mlp_g = PF(7),  *mlp_b = PF(8);
  const float* in_w1 = PF(9),  *in_b1 = PF(10);
  const float* in_w2 = PF(11), *in_b2 = PF(12);
  const float *wl[2][3], *bl[2][3], *wr[2][3];
  {
    int idx = 13;
    for (int L = 0; L < 2; ++L)
      for (int e = 0; e < 3; ++e) { wl[L][e] = PF(idx++); bl[L][e] = PF(idx++); wr[L][e] = PF(idx++); }
  }
  const float* gn_g[2] = { PF(31), PF(34) };
  const float* gn_b[2] = { PF(32), PF(35) };
  const float* gn_a[2] = { PF(33), PF(36) };
  const float* head_w1 = PF(37), *head_b1 = PF(38);
  const float* head_w2 = PF(39), *head_b2 = PF(40);
  const float* out_w1  = PF(41), *out_b1  = PF(42);
  const float* out_w2  = PF(43), *out_b2  = PF(44);
  const float* out_w3  = PF(45), *out_b3  = PF(46);
  const float* codebook = PF(47);
#undef PF

  char* ws = (char*)d_ws;
  float* B0   = (float*)(ws + 0);              // 50000x256 : h1, later JK-cat
  float* Bx   = (float*)(ws + 51200000);       // 50000x128 : current x
  float* Bagg = (float*)(ws + 76800000);       // 50000x128 : agg / head tmp A
  float* Bacc = (float*)(ws + 102400000);      // 50000x128 : conv accumulator / head tmp B
  float* cnt  = (float*)(ws + 128000000);      // 50000
  float* colsum  = (float*)(ws + 128204800);   // 128
  float* colsum2 = (float*)(ws + 128205312);   // 128
  float* loss    = (float*)(ws + 128205824);   // 1
  float* out_z   = (float*)d_out;              // 50000x32 then [1] loss

  const int N = N_NODES;

  // input block: LN -> LN -> GEMM(gelu) -> GEMM(gelu)
  mk1_ln2_kernel<<<(N + 7) / 8, 256, 0, stream>>>(x_res, ln1_g, ln1_b, mlp_g, mlp_b, Bx);
  launch_gemm(Bx, in_w1, B0, in_b1, N, 2 * HID, IN_CH, IN_CH, 2 * HID, 2 * HID, 0, 1, stream);
  launch_gemm(B0, in_w2, Bx, in_b2, N, HID, 2 * HID, 2 * HID, HID, HID, 0, 1, stream);

  // conv layers
  for (int L = 0; L < 2; ++L) {
    hipMemsetAsync(Bacc, 0, (size_t)N * HID * sizeof(float), stream);
    for (int e = 0; e < 3; ++e) {
      hipMemsetAsync(Bagg, 0, (size_t)N * HID * sizeof(float), stream);
      hipMemsetAsync(cnt, 0, (size_t)N * sizeof(float), stream);
      mk1_scatter_kernel<<<(N_EDGES + 7) / 8, 256, 0, stream>>>(edges[e], Bx, Bagg, cnt);
      mk1_agg_norm_kernel<<<2048, 256, 0, stream>>>(Bagg, cnt);
      launch_gemm(Bagg, wl[L][e], Bacc, bl[L][e], N, HID, HID, HID, HID, HID, 1, 0, stream);
      launch_gemm(Bx,   wr[L][e], Bacc, nullptr,  N, HID, HID, HID, HID, HID, 1, 0, stream);
    }
    mk1_mean_gelu_kernel<<<2048, 256, 0, stream>>>(Bacc, Bx);
    hipMemsetAsync(colsum, 0, 1024, stream);   // colsum + colsum2 contiguous
    mk1_colsum_kernel<<<100, 128, 0, stream>>>(Bx, colsum, 500);
    mk1_gn_center_kernel<<<100, 128, 0, stream>>>(Bx, colsum, gn_a[L], Bacc, colsum2, 500);
    mk1_gn_final_kernel<<<2048, 256, 0, stream>>>(Bacc, colsum2, gn_g[L], gn_b[L],
                                                  Bx, B0 + L * HID, 2 * HID);
  }

  // head: JK-cat (B0, 50000x256) -> ENC_H -> ENC_H -> (+aa) -> ENC_H -> ENC_H -> 32 tanh
  launch_gemm(B0,   head_w1, Bagg, head_b1, N, ENC_H, 2 * HID, 2 * HID, ENC_H, 128, 0, 1, stream);
  launch_gemm(Bagg, head_w2, Bacc, head_b2, N, ENC_H, ENC_H, 128, ENC_H, 128, 0, 1, stream);
  mk1_concat_aa_kernel<<<1024, 256, 0, stream>>>(aa, Bacc);
  launch_gemm(Bacc, out_w1, Bagg, out_b1, N, ENC_H, ENC_H + AA_DIM, 128, ENC_H, 128, 0, 1, stream);
  launch_gemm(Bagg, out_w2, Bacc, out_b2, N, ENC_H, ENC_H, 128, ENC_H, 128, 0, 1, stream);
  launch_gemm(Bacc, out_w3, out_z, out_b3, N, OUT_CH, ENC_H, 128, OUT_CH, OUT_CH, 0, 2, stream);

  // vector quantizer
  hipMemsetAsync(loss, 0, sizeof(float), stream);
  mk1_vq_kernel<<<(N + 255) / 256, 256, 0, stream>>>(out_z, codebook, loss);
  mk1_loss_final_kernel<<<1, 32, 0, stream>>>(loss, out_z + (size_t)N * OUT_CH);
}